// GINConv_26216480375297
// MI455X (gfx1250) — compile-verified
//
#include <hip/hip_runtime.h>

#define N_NODES 100000
#define N_EDGES 3200000
#define DIM     128
#define HS_PITCH 132   // 128 + 4 pad: row stride 132 % 64 banks -> conflict-free column reads

static_assert(N_NODES % 16 == 0, "row blocks must be exact");

typedef float v2f __attribute__((ext_vector_type(2)));
typedef float v8f __attribute__((ext_vector_type(8)));

// ---------------- zero the neighbor-accumulator workspace ----------------
__global__ void gin_zero_kernel(float4* __restrict__ p, int n4) {
    int i = blockIdx.x * blockDim.x + threadIdx.x;
    if (i < n4) {
        float4 z; z.x = 0.f; z.y = 0.f; z.z = 0.f; z.w = 0.f;
        p[i] = z;
    }
}

// ---------------- edge scatter: neigh[dst] += feat[src] ----------------
// One wave (32 lanes) per edge; each lane moves 4 floats (float4 gather,
// 4x global_atomic_add_f32 scatter). feat & neigh are L2-resident (51 MB each).
__global__ void __launch_bounds__(256)
gin_scatter_kernel(const float* __restrict__ feat,
                   const int*   __restrict__ src,
                   const int*   __restrict__ dst,
                   float*       __restrict__ neigh) {
    int gid  = blockIdx.x * blockDim.x + threadIdx.x;
    int e    = gid >> 5;
    int lane = gid & 31;
    if (e >= N_EDGES) return;
    int s = src[e];
    int d = dst[e];
    const float4 v = *reinterpret_cast<const float4*>(feat + (size_t)s * DIM + lane * 4);
    float* o = neigh + (size_t)d * DIM + lane * 4;
    atomicAdd(o + 0, v.x);
    atomicAdd(o + 1, v.y);
    atomicAdd(o + 2, v.z);
    atomicAdd(o + 3, v.w);
}

// ---------------- fused GIN MLP: out = relu(h@W1+b1)@W2 + b2 ----------------
// 256 threads = 8 waves; each wave owns one 16-column N-tile for both GEMMs.
// Weight B-fragments (128x16 slice per wave, per matrix) are hoisted into
// registers ONCE per workgroup (64+64 VGPRs), loaded straight from global
// (L2-resident). Inner loop: LDS A-fragment loads + 32 accumulating
// V_WMMA_F32_16X16X4_F32 per GEMM. LDS use is only the two 16x132 h tiles.
__global__ void __launch_bounds__(256)
gin_mlp_kernel(const float* __restrict__ feat,
               const float* __restrict__ neigh,
               const float* __restrict__ eps,
               const float* __restrict__ W1,
               const float* __restrict__ b1,
               const float* __restrict__ W2,
               const float* __restrict__ b2,
               float*       __restrict__ out,
               int nRowBlocks) {
    __shared__ float hs [16 * HS_PITCH];
    __shared__ float h1s[16 * HS_PITCH];

    const int tid  = threadIdx.x;
    const int wave = tid >> 5;
    const int lane = tid & 31;
    const int n0   = wave * 16;             // this wave's N-tile
    const int nn   = lane & 15;             // N within tile (both lane halves)
    const int kOff = (lane < 16) ? 0 : 2;   // K sub-offset for A/B fragments
    const int mA   = lane & 15;             // A-fragment row

    // Hoist loop-invariant weight B-fragments into registers (once per WG).
    // Fragment layout for V_WMMA_F32_16X16X4_F32 B (4x16):
    //   lanes 0-15 : v0 = B[k+0][n], v1 = B[k+1][n]
    //   lanes 16-31: v0 = B[k+2][n], v1 = B[k+3][n]
    v2f B1[DIM / 4];
    v2f B2[DIM / 4];
    #pragma unroll
    for (int kk = 0; kk < DIM / 4; ++kk) {
        const int k = kk * 4 + kOff;
        const float* p1 = W1 + (size_t)k * DIM + n0 + nn;
        const float* p2 = W2 + (size_t)k * DIM + n0 + nn;
        B1[kk].x = p1[0];
        B1[kk].y = p1[DIM];
        B2[kk].x = p2[0];
        B2[kk].y = p2[DIM];
    }

    const float epsv  = 1.0f + eps[0];
    const float bias1 = b1[n0 + nn];
    const float bias2 = b2[n0 + nn];

    for (int rb = blockIdx.x; rb < nRowBlocks; rb += gridDim.x) {
        const int row0 = rb * 16;

        // stage h = (1+eps)*feat + neigh for 16 rows (512 float4, 2 per thread)
        for (int i = tid; i < 16 * 32; i += 256) {
            int r  = i >> 5;
            int c4 = i & 31;
            const size_t g = (size_t)(row0 + r) * DIM + c4 * 4;
            float4 f = *reinterpret_cast<const float4*>(feat + g);
            float4 a = *reinterpret_cast<const float4*>(neigh + g);
            float* hp = hs + r * HS_PITCH + c4 * 4;
            hp[0] = epsv * f.x + a.x;
            hp[1] = epsv * f.y + a.y;
            hp[2] = epsv * f.z + a.z;
            hp[3] = epsv * f.w + a.w;
        }
        __syncthreads();

        // GEMM1: acc = h @ W1  (16x16 tile per wave, K=128 via 32 wmma)
        v8f acc = {};
        #pragma unroll
        for (int kk = 0; kk < DIM / 4; ++kk) {
            v2f a;
            const float* ap = hs + mA * HS_PITCH + kk * 4 + kOff;
            a.x = ap[0];
            a.y = ap[1];
            acc = __builtin_amdgcn_wmma_f32_16x16x4_f32(
                false, a, false, B1[kk], (short)0, acc, false, false);
        }

        // bias + ReLU -> h1 tile in LDS
        #pragma unroll
        for (int j = 0; j < 8; ++j) {
            int m = (lane < 16) ? j : (j + 8);
            float v = acc[j] + bias1;
            h1s[m * HS_PITCH + n0 + nn] = v > 0.f ? v : 0.f;
        }
        __syncthreads();

        // GEMM2: out = h1 @ W2 + b2
        v8f acc2 = {};
        #pragma unroll
        for (int kk = 0; kk < DIM / 4; ++kk) {
            v2f a;
            const float* ap = h1s + mA * HS_PITCH + kk * 4 + kOff;
            a.x = ap[0];
            a.y = ap[1];
            acc2 = __builtin_amdgcn_wmma_f32_16x16x4_f32(
                false, a, false, B2[kk], (short)0, acc2, false, false);
        }

        #pragma unroll
        for (int j = 0; j < 8; ++j) {
            int m = (lane < 16) ? j : (j + 8);
            out[(size_t)(row0 + m) * DIM + n0 + nn] = acc2[j] + bias2;
        }
        __syncthreads();  // protect hs/h1s before next row block
    }
}

extern "C" void kernel_launch(void* const* d_in, const int* in_sizes, int n_in,
                              void* d_out, int out_size, void* d_ws, size_t ws_size,
                              hipStream_t stream) {
    const float* feat = (const float*)d_in[0];
    const int*   src  = (const int*)  d_in[1];
    const int*   dst  = (const int*)  d_in[2];
    const float* eps  = (const float*)d_in[3];
    const float* W1   = (const float*)d_in[4];
    const float* b1   = (const float*)d_in[5];
    const float* W2   = (const float*)d_in[6];
    const float* b2   = (const float*)d_in[7];
    float* out   = (float*)d_out;
    float* neigh = (float*)d_ws;   // N_NODES * DIM floats (51.2 MB)

    // 1) zero neighbor accumulator
    {
        int n4 = N_NODES * DIM / 4;
        int blocks = (n4 + 255) / 256;
        gin_zero_kernel<<<blocks, 256, 0, stream>>>((float4*)neigh, n4);
    }

    // 2) edge scatter (one wave per edge)
    {
        long long threads = (long long)N_EDGES * 32;
        int blocks = (int)((threads + 255) / 256);
        gin_scatter_kernel<<<blocks, 256, 0, stream>>>(feat, src, dst, neigh);
    }

    // 3) fused MLP with f32 WMMA (register-resident weights)
    {
        int nRowBlocks = N_NODES / 16;   // 6250
        int blocks = 2048;
        gin_mlp_kernel<<<blocks, 256, 0, stream>>>(feat, neigh, eps,
                                                   W1, b1, W2, b2,
                                                   out, nRowBlocks);
    }
}